// BasicGNN_10763188043954
// MI455X (gfx1250) — compile-verified
//
#include <hip/hip_runtime.h>
#include <hip/hip_bf16.h>

typedef __attribute__((ext_vector_type(2))) float v2f;
typedef __attribute__((ext_vector_type(8))) float v8f;

#define FEAT 64

__device__ __forceinline__ void atomic_add_f32(float* p, float v) {
    __hip_atomic_fetch_add(p, v, __ATOMIC_RELAXED, __HIP_MEMORY_SCOPE_AGENT);
}

// ---------------------------------------------------------------- fill
__global__ void k_fill(float* __restrict__ p, float v, long long n) {
    long long i = (long long)blockIdx.x * blockDim.x + threadIdx.x;
    if (i < n) p[i] = v;
}

// ---------------------------------------------------------------- degree (deg starts at 1.0 for self-loop)
__global__ void k_degree(const int* __restrict__ dst, float* __restrict__ deg, int E) {
    int e = blockIdx.x * blockDim.x + threadIdx.x;
    if (e < E) atomic_add_f32(&deg[dst[e]], 1.0f);
}

// ---------------------------------------------------------------- dinv = deg^-1/2
__global__ void k_rsqrt(float* __restrict__ d, int N) {
    int i = blockIdx.x * blockDim.x + threadIdx.x;
    if (i < N) d[i] = rsqrtf(d[i]);
}

// ---------------------------------------------------------------- H = X @ W   (N x 64) * (64 x 64), fp32 WMMA
// One wave per 16-row tile; 8 waves (128 rows) per block.
// W staged in LDS pair-interleaved: Wl[(k/2)*128 + col*2 + (k&1)] so each
// B fragment (W[k][col], W[k+1][col]) is one 8B-aligned ds_load_b64 —
// no VGPR repacking between LDS load and WMMA.
__global__ __launch_bounds__(256) void k_gemm_wmma(const float* __restrict__ X,
                                                   const float* __restrict__ W,
                                                   float* __restrict__ H, int N) {
    __shared__ float Wl[FEAT * FEAT];
    for (int i = threadIdx.x; i < FEAT * FEAT; i += 256) {
        const int k   = i >> 6;
        const int col = i & 63;
        Wl[(k >> 1) * 128 + col * 2 + (k & 1)] = W[i];
    }
    __syncthreads();

    const int wave = threadIdx.x >> 5;          // 0..7
    const int lane = threadIdx.x & 31;
    const int mrow = lane & 15;                 // A row within tile / B-D column within 16-tile
    const int hi   = lane >> 4;                 // half-wave selector
    const int row0 = blockIdx.x * 128 + wave * 16;

    if (row0 >= N) return;                      // wave-uniform: EXEC stays all-1 for live waves

    int arow = row0 + mrow;                     // this lane's A row (clamped)
    if (arow > N - 1) arow = N - 1;
    const float* __restrict__ xrow = X + (size_t)arow * FEAT + (hi << 1);

    v8f acc0 = {}, acc1 = {}, acc2 = {}, acc3 = {};

    #pragma unroll
    for (int k0 = 0; k0 < FEAT; k0 += 4) {
        const int kb = k0 + (hi << 1);          // lanes 0-15: K=k0..k0+1, lanes 16-31: K=k0+2..k0+3
        // branch-free A fragment: 8-byte aligned pair -> global_load_b64
        const v2f a = *reinterpret_cast<const v2f*>(xrow + k0);

        // B fragments: contiguous interleaved pairs -> single ds_load_b64 each
        const float* __restrict__ wrow = Wl + (kb >> 1) * 128 + mrow * 2;
        const v2f b0 = *reinterpret_cast<const v2f*>(wrow +  0);
        const v2f b1 = *reinterpret_cast<const v2f*>(wrow + 32);
        const v2f b2 = *reinterpret_cast<const v2f*>(wrow + 64);
        const v2f b3 = *reinterpret_cast<const v2f*>(wrow + 96);

        acc0 = __builtin_amdgcn_wmma_f32_16x16x4_f32(false, a, false, b0, (short)0, acc0, false, false);
        acc1 = __builtin_amdgcn_wmma_f32_16x16x4_f32(false, a, false, b1, (short)0, acc1, false, false);
        acc2 = __builtin_amdgcn_wmma_f32_16x16x4_f32(false, a, false, b2, (short)0, acc2, false, false);
        acc3 = __builtin_amdgcn_wmma_f32_16x16x4_f32(false, a, false, b3, (short)0, acc3, false, false);
    }

    // D layout: VGPR j -> M = j + 8*hi, N = mrow (within each 16-col tile)
    #pragma unroll
    for (int j = 0; j < 8; ++j) {
        const int r = row0 + j + hi * 8;
        if (r < N) {
            float* __restrict__ hrow = H + (size_t)r * FEAT + mrow;
            hrow[ 0] = acc0[j];
            hrow[16] = acc1[j];
            hrow[32] = acc2[j];
            hrow[48] = acc3[j];
        }
    }
}

// ---------------------------------------------------------------- scatter: AGG[dst] += H[src]*dinv[s]*dinv[d]
// 16 threads per edge, float4 gather (global_load_b128), 4 hardware f32 atomics each.
__global__ void k_scatter(const float* __restrict__ H, float* __restrict__ AGG,
                          const int* __restrict__ src, const int* __restrict__ dst,
                          const float* __restrict__ dinv, long long total) {
    long long tid = (long long)blockIdx.x * blockDim.x + threadIdx.x;
    if (tid >= total) return;
    const int e = (int)(tid >> 4);
    const int q = (int)(tid & 15);
    const int s = src[e];
    const int d = dst[e];
    const float w = dinv[s] * dinv[d];
    const float4 hv = reinterpret_cast<const float4*>(H)[(size_t)s * 16 + q];
    float* __restrict__ out = AGG + (size_t)d * FEAT + q * 4;
    atomic_add_f32(out + 0, hv.x * w);
    atomic_add_f32(out + 1, hv.y * w);
    atomic_add_f32(out + 2, hv.z * w);
    atomic_add_f32(out + 3, hv.w * w);
}

// ---------------------------------------------------------------- AGG = relu(AGG + H*dinv^2 + b)   (self-loop + bias + relu)
__global__ void k_bias_relu_self(float* __restrict__ AGG, const float* __restrict__ H,
                                 const float* __restrict__ dinv, const float* __restrict__ b,
                                 int N) {
    long long tid = (long long)blockIdx.x * blockDim.x + threadIdx.x;
    if (tid >= (long long)N * FEAT) return;
    const int i = (int)(tid >> 6);
    const int f = (int)(tid & 63);
    const float di = dinv[i];
    float v = AGG[tid] + H[tid] * di * di + b[f];
    AGG[tid] = v > 0.0f ? v : 0.0f;
}

// ---------------------------------------------------------------- mean-pool accumulation
__global__ void k_pool(const float* __restrict__ ACT, const int* __restrict__ batch,
                       float* __restrict__ pooled, float* __restrict__ counts, int N) {
    long long tid = (long long)blockIdx.x * blockDim.x + threadIdx.x;
    if (tid >= (long long)N * FEAT) return;
    const int i = (int)(tid >> 6);
    const int f = (int)(tid & 63);
    const int g = batch[i];
    atomic_add_f32(&pooled[(size_t)g * FEAT + f], ACT[tid]);
    if (f == 0) atomic_add_f32(&counts[g], 1.0f);
}

// ---------------------------------------------------------------- head: out[g] = dot(pooled/c, Wout[:64]) + depth*Wout[64] + bout
__global__ void k_head(const float* __restrict__ pooled, const float* __restrict__ counts,
                       const float* __restrict__ depth, const float* __restrict__ Wout,
                       const float* __restrict__ bout, float* __restrict__ out, int G) {
    int g = blockIdx.x * blockDim.x + threadIdx.x;
    if (g >= G) return;
    float c = counts[g];
    c = c > 1.0f ? c : 1.0f;
    float s = 0.0f;
    #pragma unroll 8
    for (int f = 0; f < FEAT; ++f) s += pooled[(size_t)g * FEAT + f] * Wout[f];
    out[g] = s / c + depth[g] * Wout[FEAT] + bout[0];
}

// ================================================================ launch
extern "C" void kernel_launch(void* const* d_in, const int* in_sizes, int n_in,
                              void* d_out, int out_size, void* d_ws, size_t ws_size,
                              hipStream_t stream) {
    const float* x     = (const float*)d_in[0];
    const int*   eidx  = (const int*)  d_in[1];
    const int*   batch = (const int*)  d_in[2];
    const float* depth = (const float*)d_in[3];
    const float* W1    = (const float*)d_in[4];
    const float* b1    = (const float*)d_in[5];
    const float* W2    = (const float*)d_in[6];
    const float* b2    = (const float*)d_in[7];
    const float* Wout  = (const float*)d_in[8];
    const float* bout  = (const float*)d_in[9];
    float*       out   = (float*)d_out;

    const int N = in_sizes[0] / FEAT;       // 50000
    const int E = in_sizes[1] / 2;          // 800000
    const int G = in_sizes[3];              // 512

    const int* src = eidx;
    const int* dst = eidx + E;

    // workspace layout (floats)
    float* h      = (float*)d_ws;                      // N*64
    float* agg    = h      + (size_t)N * FEAT;         // N*64
    float* dinv   = agg    + (size_t)N * FEAT;         // N (deg then deg^-1/2)
    float* pooled = dinv   + N;                        // G*64
    float* counts = pooled + (size_t)G * FEAT;         // G

    const int B = 256;
    const long long nNF = (long long)N * FEAT;
    const long long nSC = (long long)E * 16;
    auto blks = [](long long n, int b) { return (unsigned)((n + b - 1) / b); };

    // init
    k_fill<<<blks(N, B), B, 0, stream>>>(dinv, 1.0f, N);
    k_fill<<<blks(nNF, B), B, 0, stream>>>(agg, 0.0f, nNF);
    k_fill<<<blks((long long)G * (FEAT + 1), B), B, 0, stream>>>(pooled, 0.0f, (long long)G * (FEAT + 1));

    // normalization
    k_degree<<<blks(E, B), B, 0, stream>>>(dst, dinv, E);
    k_rsqrt<<<blks(N, B), B, 0, stream>>>(dinv, N);

    const unsigned gemm_blocks = (unsigned)((N + 127) / 128);

    // ---- layer 1
    k_gemm_wmma<<<gemm_blocks, 256, 0, stream>>>(x, W1, h, N);
    k_scatter<<<blks(nSC, B), B, 0, stream>>>(h, agg, src, dst, dinv, nSC);
    k_bias_relu_self<<<blks(nNF, B), B, 0, stream>>>(agg, h, dinv, b1, N);

    // ---- layer 2 (agg holds act1 as GEMM input; h gets h2; then agg re-zeroed)
    k_gemm_wmma<<<gemm_blocks, 256, 0, stream>>>(agg, W2, h, N);
    k_fill<<<blks(nNF, B), B, 0, stream>>>(agg, 0.0f, nNF);
    k_scatter<<<blks(nSC, B), B, 0, stream>>>(h, agg, src, dst, dinv, nSC);
    k_bias_relu_self<<<blks(nNF, B), B, 0, stream>>>(agg, h, dinv, b2, N);

    // ---- pool + head
    k_pool<<<blks(nNF, B), B, 0, stream>>>(agg, batch, pooled, counts, N);
    k_head<<<blks(G, 64), 64, 0, stream>>>(pooled, counts, depth, Wout, bout, out, G);
}